// CrossModalAttentionFusion_22514218565666
// MI455X (gfx1250) — compile-verified
//
#include <hip/hip_runtime.h>
#include <math.h>

#define BB 512
#define SS 512
#define DD 512
#define LN_EPS 1e-5f
#define INV_SCALE 0.125f   // 1/sqrt(d_k), d_k = 64

typedef float v2f __attribute__((ext_vector_type(2)));
typedef float v8f __attribute__((ext_vector_type(8)));

__device__ __forceinline__ float wave_sum(float v) {
#pragma unroll
  for (int off = 16; off > 0; off >>= 1) v += __shfl_xor(v, off, 32);
  return v;
}

// ---------------------------------------------------------------------------
// LayerNorm over rows of length 512; one wave (32 lanes x 16 elems) per row.
// ---------------------------------------------------------------------------
__global__ __launch_bounds__(256) void ln_rows_kernel(
    const float* __restrict__ x, const float* __restrict__ g,
    const float* __restrict__ beta, float* __restrict__ out) {
  const int wave = threadIdx.x >> 5;
  const int lane = threadIdx.x & 31;
  const int row  = blockIdx.x * 8 + wave;
  const float* xr = x + (size_t)row * DD;
  float4 xs[4];
  float s = 0.f, ss = 0.f;
#pragma unroll
  for (int i = 0; i < 4; ++i) {
    xs[i] = *(const float4*)(xr + i * 128 + lane * 4);
    s  += xs[i].x + xs[i].y + xs[i].z + xs[i].w;
    ss += xs[i].x * xs[i].x + xs[i].y * xs[i].y + xs[i].z * xs[i].z + xs[i].w * xs[i].w;
  }
  s  = wave_sum(s);
  ss = wave_sum(ss);
  const float m    = s * (1.f / DD);
  const float var  = ss * (1.f / DD) - m * m;
  const float rstd = rsqrtf(var + LN_EPS);
  float* orow = out + (size_t)row * DD;
#pragma unroll
  for (int i = 0; i < 4; ++i) {
    const int d = i * 128 + lane * 4;
    float4 gv = *(const float4*)(g + d);
    float4 bv = *(const float4*)(beta + d);
    float4 o;
    o.x = (xs[i].x - m) * rstd * gv.x + bv.x;
    o.y = (xs[i].y - m) * rstd * gv.y + bv.y;
    o.z = (xs[i].z - m) * rstd * gv.z + bv.z;
    o.w = (xs[i].w - m) * rstd * gv.w + bv.w;
    *(float4*)(orow + d) = o;
  }
}

// ---------------------------------------------------------------------------
// fp32 WMMA GEMM: C[M,N] = A[M,K] * B (optionally B transposed) + bias.
// One wave per 16x16 tile; K-loop of V_WMMA_F32_16X16X4_F32.
// A frag: lane holds A[m0 + (lane&15)][k + 2*(lane>>4) + {0,1}]  (float2)
// B frag: lane holds B[k + 2*(lane>>4) + {0,1}][n0 + (lane&15)]
// C/D:    VGPR r -> C[m0 + r + 8*(lane>>4)][n0 + (lane&15)]
// ---------------------------------------------------------------------------
template <int TRANSB, int BIAS>
__global__ __launch_bounds__(32) void gemm16_kernel(
    const float* __restrict__ A, int lda, const float* __restrict__ Bm, int ldb,
    const float* __restrict__ bias, float* __restrict__ C, int ldc, int K) {
  const int lane = threadIdx.x & 31;
  const int n0 = blockIdx.x * 16;
  const int m0 = blockIdx.y * 16;
  const int mc = lane & 15;   // A row / B col / C col within tile
  const int kh = lane >> 4;   // K half-select
  v8f acc = {0.f, 0.f, 0.f, 0.f, 0.f, 0.f, 0.f, 0.f};
  const float* Arow = A + (size_t)(m0 + mc) * lda + 2 * kh;
  for (int k = 0; k < K; k += 4) {
    v2f a = *(const v2f*)(Arow + k);
    v2f b;
    if (TRANSB) {
      const float* Br = Bm + (size_t)(n0 + mc) * ldb + k + 2 * kh;
      b.x = Br[0];
      b.y = Br[1];
    } else {
      const float* Bc = Bm + (size_t)(k + 2 * kh) * ldb + n0 + mc;
      b.x = Bc[0];
      b.y = Bc[ldb];
    }
    acc = __builtin_amdgcn_wmma_f32_16x16x4_f32(false, a, false, b, (short)0,
                                                acc, false, false);
  }
  float bval = 0.f;
  if (BIAS) bval = bias[n0 + mc];
#pragma unroll
  for (int r = 0; r < 8; ++r) {
    const int row = r + 8 * kh;
    C[(size_t)(m0 + row) * ldc + n0 + mc] = acc[r] + bval;
  }
}

// ---------------------------------------------------------------------------
// cb[b] = dot(q[b], bk) * INV_SCALE  (wave per row)
// ---------------------------------------------------------------------------
__global__ __launch_bounds__(256) void rowdot_kernel(
    const float* __restrict__ q, const float* __restrict__ bk,
    float* __restrict__ cb) {
  const int wave = threadIdx.x >> 5;
  const int lane = threadIdx.x & 31;
  const int row  = blockIdx.x * 8 + wave;
  const float* qr = q + (size_t)row * DD;
  float s = 0.f;
#pragma unroll
  for (int i = 0; i < 4; ++i) {
    const int d = i * 128 + lane * 4;
    float4 a = *(const float4*)(qr + d);
    float4 b = *(const float4*)(bk + d);
    s += a.x * b.x + a.y * b.y + a.z * b.z + a.w * b.w;
  }
  s = wave_sum(s);
  if (lane == 0) cb[row] = s * INV_SCALE;
}

// ---------------------------------------------------------------------------
// Fused LN + single-query attention, flash-style single pass over price_seq.
// One block (8 waves) per batch element. Writes ctxKV[b,:] and attn[b,:].
// ---------------------------------------------------------------------------
__global__ __launch_bounds__(256) void attn_kernel(
    const float* __restrict__ price, const float* __restrict__ g,
    const float* __restrict__ beta, const float* __restrict__ qt,
    const float* __restrict__ cb, float* __restrict__ ctxKV,
    float* __restrict__ attn_out) {
  __shared__ float s_scores[SS];
  __shared__ float s_ctx[8][DD];
  __shared__ float s_wm[8];
  __shared__ float s_wl[8];
  const int b    = blockIdx.x;
  const int wave = threadIdx.x >> 5;
  const int lane = threadIdx.x & 31;

  float qv[16], gv[16], bv[16];
#pragma unroll
  for (int i = 0; i < 4; ++i) {
    const int d = i * 128 + lane * 4;
    float4 t0 = *(const float4*)(qt + (size_t)b * DD + d);
    float4 t1 = *(const float4*)(g + d);
    float4 t2 = *(const float4*)(beta + d);
    qv[4 * i] = t0.x; qv[4 * i + 1] = t0.y; qv[4 * i + 2] = t0.z; qv[4 * i + 3] = t0.w;
    gv[4 * i] = t1.x; gv[4 * i + 1] = t1.y; gv[4 * i + 2] = t1.z; gv[4 * i + 3] = t1.w;
    bv[4 * i] = t2.x; bv[4 * i + 1] = t2.y; bv[4 * i + 2] = t2.z; bv[4 * i + 3] = t2.w;
  }
  const float cbb = cb[b];

  float ctx[16];
#pragma unroll
  for (int i = 0; i < 16; ++i) ctx[i] = 0.f;
  float M = -INFINITY, L = 0.f;

  for (int s = wave; s < SS; s += 8) {
    const float* xr = price + ((size_t)b * SS + s) * DD;
    float x[16];
    float sum = 0.f, ssum = 0.f;
#pragma unroll
    for (int i = 0; i < 4; ++i) {
      float4 t = *(const float4*)(xr + i * 128 + lane * 4);
      x[4 * i] = t.x; x[4 * i + 1] = t.y; x[4 * i + 2] = t.z; x[4 * i + 3] = t.w;
      sum  += t.x + t.y + t.z + t.w;
      ssum += t.x * t.x + t.y * t.y + t.z * t.z + t.w * t.w;
    }
    sum  = wave_sum(sum);
    ssum = wave_sum(ssum);
    const float m    = sum * (1.f / DD);
    const float var  = ssum * (1.f / DD) - m * m;
    const float rstd = rsqrtf(var + LN_EPS);
    float dot = 0.f;
#pragma unroll
    for (int i = 0; i < 16; ++i) {
      x[i] = (x[i] - m) * rstd * gv[i] + bv[i];  // LN'd KV value
      dot += x[i] * qv[i];
    }
    dot = wave_sum(dot);
    const float score = dot * INV_SCALE + cbb;
    if (lane == 0) s_scores[s] = score;
    const float nM = fmaxf(M, score);
    const float c  = __expf(M - nM);      // 0 on first iter (M = -inf)
    const float w  = __expf(score - nM);
    L = L * c + w;
#pragma unroll
    for (int i = 0; i < 16; ++i) ctx[i] = ctx[i] * c + w * x[i];
    M = nM;
  }

#pragma unroll
  for (int i = 0; i < 4; ++i) {
    const int d = i * 128 + lane * 4;
    float4 t;
    t.x = ctx[4 * i]; t.y = ctx[4 * i + 1]; t.z = ctx[4 * i + 2]; t.w = ctx[4 * i + 3];
    *(float4*)(&s_ctx[wave][d]) = t;
  }
  if (lane == 0) { s_wm[wave] = M; s_wl[wave] = L; }
  __syncthreads();

  float gM = s_wm[0];
#pragma unroll
  for (int w = 1; w < 8; ++w) gM = fmaxf(gM, s_wm[w]);
  float sc[8];
  float gL = 0.f;
#pragma unroll
  for (int w = 0; w < 8; ++w) {
    sc[w] = __expf(s_wm[w] - gM);
    gL += s_wl[w] * sc[w];
  }
  const float invL = 1.f / gL;

  for (int d = threadIdx.x; d < DD; d += 256) {
    float v = 0.f;
#pragma unroll
    for (int w = 0; w < 8; ++w) v += sc[w] * s_ctx[w][d];
    ctxKV[(size_t)b * DD + d] = v * invL;
  }
  for (int s = threadIdx.x; s < SS; s += 256) {
    attn_out[(size_t)b * SS + s] = __expf(s_scores[s] - gM) * invL;
  }
}

// ---------------------------------------------------------------------------
// Gate: alpha = sigmoid([sent, ctx2] @ Wg + bg); fused = a*ctx2 + (1-a)*sent.
// K = 1024 split into two K=512 phases (sent rows, then ctx2 rows of Wg).
// ---------------------------------------------------------------------------
__global__ __launch_bounds__(32) void gate_kernel(
    const float* __restrict__ sent, const float* __restrict__ ctx2,
    const float* __restrict__ Wg, const float* __restrict__ bg,
    float* __restrict__ fused) {
  const int lane = threadIdx.x & 31;
  const int n0 = blockIdx.x * 16;
  const int m0 = blockIdx.y * 16;
  const int mc = lane & 15;
  const int kh = lane >> 4;
  v8f acc = {0.f, 0.f, 0.f, 0.f, 0.f, 0.f, 0.f, 0.f};
  const float* A0 = sent + (size_t)(m0 + mc) * DD + 2 * kh;
  const float* A1 = ctx2 + (size_t)(m0 + mc) * DD + 2 * kh;
  const float* W0 = Wg;                         // rows [0, D)
  const float* W1 = Wg + (size_t)DD * DD;       // rows [D, 2D)
  for (int k = 0; k < DD; k += 4) {
    v2f a = *(const v2f*)(A0 + k);
    v2f bvec;
    const float* Bc = W0 + (size_t)(k + 2 * kh) * DD + n0 + mc;
    bvec.x = Bc[0];
    bvec.y = Bc[DD];
    acc = __builtin_amdgcn_wmma_f32_16x16x4_f32(false, a, false, bvec, (short)0,
                                                acc, false, false);
  }
  for (int k = 0; k < DD; k += 4) {
    v2f a = *(const v2f*)(A1 + k);
    v2f bvec;
    const float* Bc = W1 + (size_t)(k + 2 * kh) * DD + n0 + mc;
    bvec.x = Bc[0];
    bvec.y = Bc[DD];
    acc = __builtin_amdgcn_wmma_f32_16x16x4_f32(false, a, false, bvec, (short)0,
                                                acc, false, false);
  }
  const float bgv = bg[n0 + mc];
#pragma unroll
  for (int r = 0; r < 8; ++r) {
    const int m = m0 + r + 8 * kh;
    const int n = n0 + mc;
    const float pre   = acc[r] + bgv;
    const float alpha = 1.f / (1.f + __expf(-pre));
    const float cv = ctx2[(size_t)m * DD + n];
    const float sv = sent[(size_t)m * DD + n];
    fused[(size_t)m * DD + n] = alpha * cv + (1.f - alpha) * sv;
  }
}

// ---------------------------------------------------------------------------
extern "C" void kernel_launch(void* const* d_in, const int* in_sizes, int n_in,
                              void* d_out, int out_size, void* d_ws,
                              size_t ws_size, hipStream_t stream) {
  (void)in_sizes; (void)n_in; (void)out_size; (void)ws_size;
  const float* sent   = (const float*)d_in[0];
  const float* price  = (const float*)d_in[1];
  const float* Wq     = (const float*)d_in[2];
  const float* bq     = (const float*)d_in[3];
  const float* Wk     = (const float*)d_in[4];
  const float* bk     = (const float*)d_in[5];
  const float* Wv     = (const float*)d_in[6];
  const float* bv     = (const float*)d_in[7];
  const float* Wo     = (const float*)d_in[8];
  const float* bo     = (const float*)d_in[9];
  const float* Wg     = (const float*)d_in[10];
  const float* bg     = (const float*)d_in[11];
  const float* g_q    = (const float*)d_in[12];
  const float* beta_q = (const float*)d_in[13];
  const float* g_kv   = (const float*)d_in[14];
  const float* beta_kv= (const float*)d_in[15];

  float* out   = (float*)d_out;
  float* fused = out;                          // [B, D]
  float* attn  = out + (size_t)BB * DD;        // [B, S]

  float* ws    = (float*)d_ws;
  const size_t BD = (size_t)BB * DD;
  float* Q      = ws;            // LN'd sentiment  [B, D]
  float* qp     = Q + BD;        // q = Q@Wq + bq   [B, D]
  float* qt     = qp + BD;       // q @ Wk^T        [B, D]
  float* ctxKV  = qt + BD;       // softmax-weighted LN(KV)  [B, D]
  float* ctxV   = ctxKV + BD;    // ctxKV@Wv + bv   [B, D]
  float* ctx2   = ctxV + BD;     // ctxV@Wo + bo    [B, D]
  float* cb     = ctx2 + BD;     // q.bk / SCALE    [B]

  dim3 g16(DD / 16, BB / 16);

  ln_rows_kernel<<<BB / 8, 256, 0, stream>>>(sent, g_q, beta_q, Q);
  gemm16_kernel<0, 1><<<g16, 32, 0, stream>>>(Q, DD, Wq, DD, bq, qp, DD, DD);
  gemm16_kernel<1, 0><<<g16, 32, 0, stream>>>(qp, DD, Wk, DD, nullptr, qt, DD, DD);
  rowdot_kernel<<<BB / 8, 256, 0, stream>>>(qp, bk, cb);
  attn_kernel<<<BB, 256, 0, stream>>>(price, g_kv, beta_kv, qt, cb, ctxKV, attn);
  gemm16_kernel<0, 1><<<g16, 32, 0, stream>>>(ctxKV, DD, Wv, DD, bv, ctxV, DD, DD);
  gemm16_kernel<0, 1><<<g16, 32, 0, stream>>>(ctxV, DD, Wo, DD, bo, ctx2, DD, DD);
  gate_kernel<<<g16, 32, 0, stream>>>(sent, ctx2, Wg, bg, fused);
}